// GeometricDistanceHead_23922967838823
// MI455X (gfx1250) — compile-verified
//
#include <hip/hip_runtime.h>

#define KDIM 4096
#define FDIM 64
#define BDIM 4
#define BKTOT (BDIM * KDIM)

typedef float v8f __attribute__((ext_vector_type(8)));
typedef float v2f __attribute__((ext_vector_type(2)));

__device__ __forceinline__ float reluf(float x) { return fmaxf(x, 0.0f); }
__device__ __forceinline__ float sqf(float x) { return x * x; }
__device__ __forceinline__ void atomic_max_pos(float* p, float v) {
    // valid for non-negative IEEE floats: int bit pattern is monotonic
    atomicMax((int*)p, __float_as_int(v));
}
__device__ __forceinline__ float softplusf(float x) {
    return fmaxf(x, 0.0f) + log1pf(__expf(-fabsf(x)));
}

// ---------------------------------------------------------------------------
// Phase 0: squared norms of features + zero-init accumulators (every call!)
// ---------------------------------------------------------------------------
__global__ __launch_bounds__(256) void p0_init(
    const float* __restrict__ pc, const float* __restrict__ img,
    float* __restrict__ sqp, float* __restrict__ sqi,
    float* __restrict__ rmax, float* __restrict__ cmax,
    float* __restrict__ csp, float* __restrict__ csn) {
    int i = blockIdx.x * 256 + threadIdx.x;      // i = b*K + n
    int b = i >> 12;
    int n = i & (KDIM - 1);
    const float* pb = pc + (size_t)b * FDIM * KDIM + n;
    const float* ib = img + (size_t)b * FDIM * KDIM + n;
    float sp = 0.0f, si = 0.0f;
#pragma unroll 8
    for (int f = 0; f < FDIM; ++f) {
        float a = pb[(size_t)f * KDIM]; sp += a * a;
        float c = ib[(size_t)f * KDIM]; si += c * c;
    }
    sqp[i] = sp; sqi[i] = si;
    rmax[i] = 0.0f; cmax[i] = 0.0f; csp[i] = 0.0f; csn[i] = 0.0f;
}

// ---------------------------------------------------------------------------
// Phase A: fp32 WMMA GEMM -> dists, + row/col maxima of positive logits
// block = 256 threads (8 waves), block tile 128x128, wave tile 32x64
// ---------------------------------------------------------------------------
__global__ __launch_bounds__(256) void pA_gemm(
    const float* __restrict__ img, const float* __restrict__ pc,
    const float* __restrict__ dmap, float* __restrict__ dists,
    const float* __restrict__ sqp, const float* __restrict__ sqi,
    float* __restrict__ rmax, float* __restrict__ cmax) {
    const int tid  = threadIdx.x;
    const int lane = tid & 31;
    const int w    = tid >> 5;
    const int l15  = lane & 15;
    const int kh   = lane >> 4;            // half-wave selector
    const int b    = blockIdx.z;
    const int nb   = blockIdx.y * 128 + (w >> 1) * 32;   // pc rows
    const int mb   = blockIdx.x * 128 + (w & 1) * 64;    // img cols
    const int bk   = b * KDIM;

    const float* pcb = pc  + (size_t)b * FDIM * KDIM;
    const float* imb = img + (size_t)b * FDIM * KDIM;

    v8f acc[2][4];
#pragma unroll
    for (int i = 0; i < 2; ++i)
#pragma unroll
        for (int j = 0; j < 4; ++j) acc[i][j] = 0.0f;

#pragma unroll
    for (int f0 = 0; f0 < FDIM; f0 += 4) {
        const int fA = f0 + (kh << 1);     // this half-wave's K rows {fA, fA+1}
        v2f afr[2], bfr[4];
#pragma unroll
        for (int ti = 0; ti < 2; ++ti) {   // A = pc^T : A[n][f] = pc[b][f][n]
            int n = nb + ti * 16 + l15;
            afr[ti].x = pcb[(size_t)(fA + 0) * KDIM + n];
            afr[ti].y = pcb[(size_t)(fA + 1) * KDIM + n];
        }
#pragma unroll
        for (int tj = 0; tj < 4; ++tj) {   // B = img : B[f][m]
            int m = mb + tj * 16 + l15;
            bfr[tj].x = imb[(size_t)(fA + 0) * KDIM + m];
            bfr[tj].y = imb[(size_t)(fA + 1) * KDIM + m];
        }
#pragma unroll
        for (int ti = 0; ti < 2; ++ti)
#pragma unroll
            for (int tj = 0; tj < 4; ++tj)
                acc[ti][tj] = __builtin_amdgcn_wmma_f32_16x16x4_f32(
                    false, afr[ti], false, bfr[tj], (short)0, acc[ti][tj],
                    false, false);
    }

    // ---- epilogue: d = sqrt(|pc|^2 + |img|^2 - 2*cross), store, track maxes
    float sm[4];
#pragma unroll
    for (int tj = 0; tj < 4; ++tj) sm[tj] = sqi[bk + mb + tj * 16 + l15];
    const float* dmb = dmap + (size_t)bk * KDIM;
    float* dob = dists + (size_t)bk * KDIM;

    float cmv[4] = {0.0f, 0.0f, 0.0f, 0.0f};
#pragma unroll
    for (int ti = 0; ti < 2; ++ti) {
#pragma unroll
        for (int r = 0; r < 8; ++r) {
            const int n = nb + ti * 16 + r + (kh << 3);
            const float sn = sqp[bk + n];
            const size_t ro = (size_t)n * KDIM;
            float rmv = 0.0f;
#pragma unroll
            for (int tj = 0; tj < 4; ++tj) {
                const int m = mb + tj * 16 + l15;
                float c  = acc[ti][tj][r];
                float d2 = sn + sm[tj] - 2.0f * c;
                float d  = sqrtf(fmaxf(d2, 1e-12f));
                dob[ro + m] = d;
                float dm = dmb[ro + m];
                float pl = (dm <= 1.0f) ? 10.0f * sqf(reluf(d - 0.1f)) : 0.0f;
                rmv = fmaxf(rmv, pl);
                cmv[tj] = fmaxf(cmv[tj], pl);
            }
            // reduce row max across 16 lanes (stays within half-wave)
#pragma unroll
            for (int msk = 8; msk >= 1; msk >>= 1)
                rmv = fmaxf(rmv, __shfl_xor(rmv, msk, 32));
            if (l15 == 0) atomic_max_pos(&rmax[bk + n], rmv);
        }
    }
#pragma unroll
    for (int tj = 0; tj < 4; ++tj) {       // same column in both half-waves
        float cv = fmaxf(cmv[tj], __shfl_xor(cmv[tj], 16, 32));
        if (kh == 0) atomic_max_pos(&cmax[bk + mb + tj * 16 + l15], cv);
    }
}

// ---------------------------------------------------------------------------
// Phase B: streaming LSE sums. Block = 32 rows x full 4096 cols of batch b.
// ---------------------------------------------------------------------------
__global__ __launch_bounds__(256) void pB_sums(
    const float* __restrict__ dists, const float* __restrict__ dmap,
    const float* __restrict__ rmax, const float* __restrict__ cmax,
    float* __restrict__ rsp, float* __restrict__ rsn,
    float* __restrict__ csp, float* __restrict__ csn) {
    __shared__ float s_cmax[KDIM];
    __shared__ float s_csp[KDIM];
    __shared__ float s_csn[KDIM];
    const int tid = threadIdx.x;
    const int b   = blockIdx.y;
    const int bk  = b * KDIM;
    const int nb  = blockIdx.x * 32;

    for (int i = tid; i < KDIM; i += 256) {
        s_cmax[i] = cmax[bk + i];
        s_csp[i] = 0.0f;
        s_csn[i] = 0.0f;
    }
    __syncthreads();

    const int w = tid >> 5, lane = tid & 31;
    const int n0 = nb + w * 4;                 // 4 wave-exclusive rows
    float rm[4], rp[4] = {0, 0, 0, 0}, rn[4] = {0, 0, 0, 0};
    size_t roff[4];
#pragma unroll
    for (int i = 0; i < 4; ++i) {
        rm[i]   = rmax[bk + n0 + i];
        roff[i] = (size_t)(bk + n0 + i) * KDIM;
    }

    for (int m = lane; m < KDIM; m += 32) {
        if (m + 256 < KDIM) {
            __builtin_prefetch(&dists[roff[0] + m + 256], 0, 0);
            __builtin_prefetch(&dmap[roff[0] + m + 256], 0, 0);
        }
        const float cm = s_cmax[m];
        float cps = 0.0f, cns = 0.0f;
#pragma unroll
        for (int i = 0; i < 4; ++i) {
            float d  = dists[roff[i] + m];
            float dm = dmap[roff[i] + m];
            bool pos = (dm <= 1.0f);
            float pl = pos ? 10.0f * sqf(reluf(d - 0.1f)) : 0.0f;
            float nl = pos ? 0.0f : 10.0f * sqf(reluf(1.4f - d));
            float en = __expf(nl);
            rp[i] += __expf(pl - rm[i]);
            rn[i] += en;
            cps += __expf(pl - cm);
            cns += en;
        }
        atomicAdd(&s_csp[m], cps);
        atomicAdd(&s_csn[m], cns);
    }

#pragma unroll
    for (int i = 0; i < 4; ++i) {
#pragma unroll
        for (int msk = 16; msk >= 1; msk >>= 1) {
            rp[i] += __shfl_xor(rp[i], msk, 32);
            rn[i] += __shfl_xor(rn[i], msk, 32);
        }
        if (lane == 0) {               // rows are block/wave-exclusive
            rsp[bk + n0 + i] = rp[i];
            rsn[bk + n0 + i] = rn[i];
        }
    }
    __syncthreads();
    for (int m = tid; m < KDIM; m += 256) {
        atomicAdd(&csp[bk + m], s_csp[m]);
        atomicAdd(&csn[bk + m], s_csn[m]);
    }
}

// ---------------------------------------------------------------------------
// Phase C: final scalar loss
// ---------------------------------------------------------------------------
__global__ __launch_bounds__(256) void pC_loss(
    const float* __restrict__ rmax, const float* __restrict__ cmax,
    const float* __restrict__ rsp, const float* __restrict__ rsn,
    const float* __restrict__ csp, const float* __restrict__ csn,
    float* __restrict__ loss) {
    __shared__ float red[256];
    const int tid = threadIdx.x;
    float acc = 0.0f;
    for (int i = tid; i < BKTOT; i += 256) {
        float lse_pos_row = rmax[i] + __logf(rsp[i]);
        float lse_neg_row = __logf(rsn[i]);
        float lse_pos_col = cmax[i] + __logf(csp[i]);
        float lse_neg_col = __logf(csn[i]);
        acc += softplusf(lse_pos_row + lse_neg_row) +
               softplusf(lse_pos_col + lse_neg_col);
    }
    red[tid] = acc;
    __syncthreads();
    for (int s = 128; s > 0; s >>= 1) {
        if (tid < s) red[tid] += red[tid + s];
        __syncthreads();
    }
    if (tid == 0) loss[0] = red[0] * (0.1f / (float)BKTOT);
}

// ---------------------------------------------------------------------------
extern "C" void kernel_launch(void* const* d_in, const int* in_sizes, int n_in,
                              void* d_out, int out_size, void* d_ws, size_t ws_size,
                              hipStream_t stream) {
    (void)in_sizes; (void)n_in; (void)out_size; (void)ws_size;
    const float* img  = (const float*)d_in[0];   // [B,F,K]
    const float* pc   = (const float*)d_in[1];   // [B,F,K]
    const float* dmap = (const float*)d_in[2];   // [B,K,K]
    float* loss  = (float*)d_out;                // out[0] = loss
    float* dists = loss + 1;                     // out[1..] = dists [B,K,K]

    // workspace layout: 8 arrays of B*K floats (512 KB total)
    float* ws   = (float*)d_ws;
    float* sqp  = ws + 0 * BKTOT;
    float* sqi  = ws + 1 * BKTOT;
    float* rmax = ws + 2 * BKTOT;
    float* cmax = ws + 3 * BKTOT;
    float* rsp  = ws + 4 * BKTOT;
    float* rsn  = ws + 5 * BKTOT;
    float* csp  = ws + 6 * BKTOT;
    float* csn  = ws + 7 * BKTOT;

    p0_init<<<BKTOT / 256, 256, 0, stream>>>(pc, img, sqp, sqi, rmax, cmax, csp, csn);
    pA_gemm<<<dim3(KDIM / 128, KDIM / 128, BDIM), 256, 0, stream>>>(
        img, pc, dmap, dists, sqp, sqi, rmax, cmax);
    pB_sums<<<dim3(KDIM / 32, BDIM), 256, 0, stream>>>(
        dists, dmap, rmax, cmax, rsp, rsn, csp, csn);
    pC_loss<<<1, 256, 0, stream>>>(rmax, cmax, rsp, rsn, csp, csn, loss);
}